// ParaphraseGenerator_86955907874934
// MI455X (gfx1250) — compile-verified
//
#include <hip/hip_runtime.h>
#include <math.h>

typedef __attribute__((ext_vector_type(2))) float v2f;
typedef __attribute__((ext_vector_type(8))) float v8f;

#define S_LEN 32
#define BATCH 64
#define MROWS (S_LEN * BATCH)   /* 2048 */
#define HID   512
#define VOCAB 30000
#define THRV  1e-6f

#define KC         64           /* K-chunk staged in LDS                       */
#define LDS_STRIDE 68           /* KC+4 floats: 272B row stride                */
                                /* -> b128 aligned, bank-conflict-free A reads */

__device__ __forceinline__ float sigmoidf_(float x) { return 1.0f / (1.0f + __expf(-x)); }

// gfx1250 async copy: global -> LDS, 16B packet, tracked by ASYNCcnt.
__device__ __forceinline__ void async_b128_to_lds(uint32_t lds_byte_off, const void* gptr)
{
    asm volatile("global_load_async_to_lds_b128 %0, %1, off"
                 :: "v"(lds_byte_off), "v"(gptr) : "memory");
}
// wait until ALL async packets have landed
__device__ __forceinline__ void wait_async0() { asm volatile("s_wait_asynccnt 0" ::: "memory"); }
// wait until only the 2 most recent packets (next chunk) remain in flight;
// async loads complete in order -> previous chunk's packets have landed.
__device__ __forceinline__ void wait_async2() { asm volatile("s_wait_asynccnt 2" ::: "memory"); }

// ---------------------------------------------------------------------------
// WMMA fp32 GEMM: C[M,N] = act(A[M,K] @ W[N,K]^T + bias[N])
// Block = 4 waves sharing one 16-row M tile. A chunks are staged into LDS by
// global_load_async_to_lds_b128 with double buffering (async engine overlaps
// the WMMA consumption of the previous chunk). Each wave owns one 16-col
// N tile and streams its W rows from global (L2-resident).
// EXPA: apply exp() to A elements on use (fused softmax-probs consumer).
// ACT : 0 = none, 1 = threshold(1e-6).
// ---------------------------------------------------------------------------
template <int ACT, int EXPA>
__global__ void wmma_gemm_kernel(const float* __restrict__ A,
                                 const float* __restrict__ W,
                                 const float* __restrict__ bias,
                                 float* __restrict__ C,
                                 int M, int N, int K)
{
    __shared__ float tileA[2][16 * LDS_STRIDE];

    const int lane  = threadIdx.x & 31;
    const int wv    = threadIdx.x >> 5;
    const int tn    = blockIdx.x * 4 + wv;     // N tile (wave-uniform)
    const int tm    = blockIdx.y;              // M tile (block-uniform)
    const bool valid = (tn * 16 < N);          // wave-uniform store predicate
    const int tnc   = valid ? tn : 0;          // clamp for safe loads

    const int lidx  = lane & 15;
    const int lhalf = lane >> 4;               // 0: K={k,k+1}  1: K={k+2,k+3}

    const float* __restrict__ Abase = A + (size_t)(tm * 16) * K;
    const float* __restrict__ Wrow  = W + (size_t)(tnc * 16 + lidx) * K;

    // full-chunk staging: exactly 2 b128 packets per thread (256 quads/chunk)
    const int q0row = threadIdx.x >> 4,        const_q0c4 = 0; (void)const_q0c4;
    const int q0c4  = threadIdx.x & 15;
    const int q1row = (threadIdx.x + 128) >> 4;
    const int q1c4  = q0c4;
    const uint32_t l0 = (uint32_t)(q0row * LDS_STRIDE + q0c4 * 4) * 4u;
    const uint32_t l1 = (uint32_t)(q1row * LDS_STRIDE + q1c4 * 4) * 4u;

    const int nfull = K / KC;
    const int tail  = K - nfull * KC;          // multiple of 4 (48 for K=30000)

    v8f acc = {0.f, 0.f, 0.f, 0.f, 0.f, 0.f, 0.f, 0.f};

    // prologue: stage chunk 0 into buffer 0
    if (nfull > 0) {
        const uint32_t lb = (uint32_t)(uintptr_t)&tileA[0][0];
        async_b128_to_lds(lb + l0, Abase + (size_t)q0row * K + q0c4 * 4);
        async_b128_to_lds(lb + l1, Abase + (size_t)q1row * K + q1c4 * 4);
    }

    for (int c = 0; c < nfull; ++c) {
        const int kc = c * KC;
        if (c + 1 < nfull) {       // issue next chunk, then wait for current
            const uint32_t lb = (uint32_t)(uintptr_t)&tileA[(c + 1) & 1][0];
            const int kn = kc + KC;
            async_b128_to_lds(lb + l0, Abase + (size_t)q0row * K + kn + q0c4 * 4);
            async_b128_to_lds(lb + l1, Abase + (size_t)q1row * K + kn + q1c4 * 4);
            wait_async2();
        } else {
            wait_async0();
        }
        __syncthreads();

        const float* __restrict__ tA = &tileA[c & 1][0] + lidx * LDS_STRIDE;
#pragma unroll
        for (int kk = 0; kk < KC; kk += 4) {
            const int ka = kk + lhalf * 2;
            v2f a, b;
            a.x = tA[ka];
            a.y = tA[ka + 1];
            if (EXPA) { a.x = __expf(a.x); a.y = __expf(a.y); }
            const int kg = kc + ka;
            b.x = Wrow[kg];
            b.y = Wrow[kg + 1];
            acc = __builtin_amdgcn_wmma_f32_16x16x4_f32(false, a, false, b,
                                                        (short)0, acc, false, false);
        }
        __syncthreads();
    }

    if (tail > 0) {                // runtime tail chunk (K % 64, e.g. 48)
        const int kc  = nfull * KC;
        const int qpr = tail >> 2;
        const int quads = 16 * qpr;
        const uint32_t lb = (uint32_t)(uintptr_t)&tileA[nfull & 1][0];
        for (int q = threadIdx.x; q < quads; q += blockDim.x) {
            const int row = q / qpr;
            const int c4  = q - row * qpr;
            async_b128_to_lds(lb + (uint32_t)(row * LDS_STRIDE + c4 * 4) * 4u,
                              Abase + (size_t)row * K + kc + c4 * 4);
        }
        wait_async0();
        __syncthreads();
        const float* __restrict__ tA = &tileA[nfull & 1][0] + lidx * LDS_STRIDE;
        for (int kk = 0; kk < tail; kk += 4) {
            const int ka = kk + lhalf * 2;
            v2f a, b;
            a.x = tA[ka];
            a.y = tA[ka + 1];
            if (EXPA) { a.x = __expf(a.x); a.y = __expf(a.y); }
            const int kg = kc + ka;
            b.x = Wrow[kg];
            b.y = Wrow[kg + 1];
            acc = __builtin_amdgcn_wmma_f32_16x16x4_f32(false, a, false, b,
                                                        (short)0, acc, false, false);
        }
    }

    if (valid) {
        const int   ccol  = tn * 16 + lidx;
        const float bv    = bias ? bias[ccol] : 0.0f;
        const int   crow0 = tm * 16 + lhalf * 8;
#pragma unroll
        for (int i = 0; i < 8; ++i) {
            float v = acc[i] + bv;
            if (ACT == 1) v = (v > THRV) ? v : 0.0f;
            C[(size_t)(crow0 + i) * N + ccol] = v;
        }
    }
}

// one-hot(idx) @ W1.T  ==  column gather of W1 (H,V), + bias + threshold
__global__ void gather_col_bias_thresh(const int* __restrict__ idx,   // (S*B)
                                       const float* __restrict__ W1,  // (H,V)
                                       const float* __restrict__ b1,  // (H)
                                       float* __restrict__ out,       // (S*B,H)
                                       int H, int V)
{
    const int sb = blockIdx.x;
    const int h  = blockIdx.y * blockDim.x + threadIdx.x;
    if (h >= H) return;
    const int c = idx[sb];
    float v = W1[(size_t)h * V + c] + b1[h];
    out[(size_t)sb * H + h] = (v > THRV) ? v : 0.0f;
}

// dec_in[0]=enc_phrase, dec_in[s]=gen_emb_tab[sim[s-1]]
__global__ void build_decin(const int* __restrict__ sim,    // (S,B)
                            const float* __restrict__ encp, // (B,E)
                            const float* __restrict__ tab,  // (V,E)
                            float* __restrict__ out)        // (S,B,E)
{
    const int e  = blockIdx.y * blockDim.x + threadIdx.x;   // 0..511
    const int sb = blockIdx.x;                              // 0..2047
    const int s = sb / BATCH, b = sb % BATCH;
    float v;
    if (s == 0) v = encp[b * HID + e];
    else        v = tab[(size_t)sim[(s - 1) * BATCH + b] * HID + e];
    out[(size_t)sb * HID + e] = v;
}

__global__ void gru_gate(const float* __restrict__ xw_t,  // (B,3R)
                         const float* __restrict__ hg,    // (B,3R)
                         const float* __restrict__ hprev, // (B,R)
                         float* __restrict__ hnew,        // (B,R)
                         int R)
{
    const int t = blockIdx.x * blockDim.x + threadIdx.x;
    if (t >= BATCH * R) return;
    const int b = t / R, j = t % R;
    const float* x = xw_t + (size_t)b * 3 * R;
    const float* g = hg   + (size_t)b * 3 * R;
    const float r = sigmoidf_(x[j] + g[j]);
    const float z = sigmoidf_(x[R + j] + g[R + j]);
    const float n = tanhf(x[2 * R + j] + r * g[2 * R + j]);
    hnew[t] = (1.0f - z) * n + z * hprev[t];
}

__global__ void lstm_gate(const float* __restrict__ xw_t, // (B,4G)
                          const float* __restrict__ hg,   // (B,4G)
                          float* __restrict__ c,          // (B,G) in/out
                          float* __restrict__ hout,       // (B,G) (= hs slice)
                          int G)
{
    const int t = blockIdx.x * blockDim.x + threadIdx.x;
    if (t >= BATCH * G) return;
    const int b = t / G, j = t % G;
    const float* x = xw_t + (size_t)b * 4 * G;
    const float* g = hg   + (size_t)b * 4 * G;
    const float gi = x[j]         + g[j];
    const float gf = x[G + j]     + g[G + j];
    const float gg = x[2 * G + j] + g[2 * G + j];
    const float go = x[3 * G + j] + g[3 * G + j];
    const float cn = sigmoidf_(gf) * c[t] + sigmoidf_(gi) * tanhf(gg);
    c[t] = cn;
    hout[t] = sigmoidf_(go) * tanhf(cn);
}

// in-place log-softmax over rows of length N (one block per row)
__global__ void log_softmax_rows(float* __restrict__ X, int N)
{
    __shared__ float red[256];
    float* row = X + (size_t)blockIdx.x * N;
    float m = -INFINITY;
    for (int i = threadIdx.x; i < N; i += blockDim.x) m = fmaxf(m, row[i]);
    red[threadIdx.x] = m; __syncthreads();
    for (int s = blockDim.x >> 1; s > 0; s >>= 1) {
        if (threadIdx.x < s) red[threadIdx.x] = fmaxf(red[threadIdx.x], red[threadIdx.x + s]);
        __syncthreads();
    }
    m = red[0]; __syncthreads();
    float sum = 0.f;
    for (int i = threadIdx.x; i < N; i += blockDim.x) sum += __expf(row[i] - m);
    red[threadIdx.x] = sum; __syncthreads();
    for (int s = blockDim.x >> 1; s > 0; s >>= 1) {
        if (threadIdx.x < s) red[threadIdx.x] += red[threadIdx.x + s];
        __syncthreads();
    }
    const float lse = m + logf(red[0]);
    for (int i = threadIdx.x; i < N; i += blockDim.x) row[i] -= lse;
}

// ---------------------------------------------------------------------------
static void gemm(const float* A, const float* W, const float* b, float* C,
                 int M, int N, int K, int act, int expa, hipStream_t st)
{
    dim3 grid((N / 16 + 3) / 4, M / 16);
    dim3 blk(128);
    if (expa)           wmma_gemm_kernel<1, 1><<<grid, blk, 0, st>>>(A, W, b, C, M, N, K);
    else if (act == 1)  wmma_gemm_kernel<1, 0><<<grid, blk, 0, st>>>(A, W, b, C, M, N, K);
    else                wmma_gemm_kernel<0, 0><<<grid, blk, 0, st>>>(A, W, b, C, M, N, K);
}

extern "C" void kernel_launch(void* const* d_in, const int* in_sizes, int n_in,
                              void* d_out, int out_size, void* d_ws, size_t ws_size,
                              hipStream_t stream)
{
    const int*   phrase    = (const int*)  d_in[0];
    const int*   sim       = (const int*)  d_in[1];
    const float* emb_W1    = (const float*)d_in[2];
    const float* emb_b1    = (const float*)d_in[3];
    const float* emb_W2    = (const float*)d_in[4];
    const float* emb_b2    = (const float*)d_in[5];
    const float* enc_Wih   = (const float*)d_in[6];
    const float* enc_Whh   = (const float*)d_in[7];
    const float* enc_bih   = (const float*)d_in[8];
    const float* enc_bhh   = (const float*)d_in[9];
    const float* enc_lin_W = (const float*)d_in[10];
    const float* enc_lin_b = (const float*)d_in[11];
    const float* gen_tab   = (const float*)d_in[12];
    const float* lstm_Wih  = (const float*)d_in[13];
    const float* lstm_Whh  = (const float*)d_in[14];
    const float* lstm_bih  = (const float*)d_in[15];
    const float* lstm_bhh  = (const float*)d_in[16];
    const float* gen_lin_W = (const float*)d_in[17];
    const float* gen_lin_b = (const float*)d_in[18];
    const float* dis_W1    = (const float*)d_in[19];
    const float* dis_b1    = (const float*)d_in[20];
    const float* dis_W2    = (const float*)d_in[21];
    const float* dis_b2    = (const float*)d_in[22];
    const float* dis_Wih   = (const float*)d_in[23];
    const float* dis_Whh   = (const float*)d_in[24];
    const float* dis_bih   = (const float*)d_in[25];
    const float* dis_bhh   = (const float*)d_in[26];
    const float* dis_lin_W = (const float*)d_in[27];
    const float* dis_lin_b = (const float*)d_in[28];

    float* out = (float*)d_out;                       // (S,B,V) logits -> logsoftmax
    float* enc_out_dst = out + (size_t)MROWS * VOCAB; // (B,ENC)
    float* enc_sim_dst = enc_out_dst + BATCH * HID;   // (B,ENC)

    // workspace layout (floats), ~43 MB total
    float* ws   = (float*)d_ws;
    float* e1   = ws;                                  // 2048*512
    float* e2   = e1  + (size_t)MROWS * HID;           // 2048*512
    float* xw3  = e2  + (size_t)MROWS * HID;           // 2048*1536
    float* xw4  = xw3 + (size_t)MROWS * 3 * HID;       // 2048*2048
    float* hs   = xw4 + (size_t)MROWS * 4 * HID;       // 2048*512
    float* hA   = hs  + (size_t)MROWS * HID;           // 64*512
    float* hB   = hA  + BATCH * HID;                   // 64*512
    float* cbuf = hB  + BATCH * HID;                   // 64*512
    float* hg   = cbuf + BATCH * HID;                  // 64*2048

    const dim3 gblk(256);
    const dim3 ggrid(MROWS, HID / 256);
    const int  gate_blocks = (BATCH * HID + 255) / 256;

    // ===================== encoder =====================
    gather_col_bias_thresh<<<ggrid, gblk, 0, stream>>>(phrase, emb_W1, emb_b1, e1, HID, VOCAB);
    gemm(e1, emb_W2, emb_b2, e2, MROWS, HID, HID, 1, 0, stream);        // e2 = thresh
    gemm(e2, enc_Wih, enc_bih, xw3, MROWS, 3 * HID, HID, 0, 0, stream); // xw
    hipMemsetAsync(hA, 0, BATCH * HID * sizeof(float), stream);
    {
        float* hc = hA; float* hn = hB;
        for (int s = 0; s < S_LEN; ++s) {
            gemm(hc, enc_Whh, enc_bhh, hg, BATCH, 3 * HID, HID, 0, 0, stream);
            gru_gate<<<gate_blocks, 256, 0, stream>>>(xw3 + (size_t)s * BATCH * 3 * HID, hg, hc, hn, HID);
            float* t = hc; hc = hn; hn = t;
        }
        gemm(hc, enc_lin_W, enc_lin_b, e1, BATCH, HID, HID, 0, 0, stream); // enc_phrase -> e1[0:32768]
    }
    float* enc_phrase = e1;  // consumed by build_decin before e1 is reused

    // ===================== generator (LSTM) =====================
    build_decin<<<dim3(MROWS, HID / 256), gblk, 0, stream>>>(sim, enc_phrase, gen_tab, e2); // dec_in -> e2
    gemm(e2, lstm_Wih, lstm_bih, xw4, MROWS, 4 * HID, HID, 0, 0, stream);
    hipMemsetAsync(hA, 0, BATCH * HID * sizeof(float), stream);
    hipMemsetAsync(cbuf, 0, BATCH * HID * sizeof(float), stream);
    {
        float* hprev = hA;
        for (int s = 0; s < S_LEN; ++s) {
            gemm(hprev, lstm_Whh, lstm_bhh, hg, BATCH, 4 * HID, HID, 0, 0, stream);
            float* hout = hs + (size_t)s * BATCH * HID;
            lstm_gate<<<gate_blocks, 256, 0, stream>>>(xw4 + (size_t)s * BATCH * 4 * HID, hg, cbuf, hout, HID);
            hprev = hout;
        }
    }
    // logits straight into d_out, then in-place log-softmax
    gemm(hs, gen_lin_W, gen_lin_b, out, MROWS, VOCAB, HID, 0, 0, stream);
    log_softmax_rows<<<MROWS, 256, 0, stream>>>(out, VOCAB);

    // ===================== discriminator on sim_phrase =====================
    gather_col_bias_thresh<<<ggrid, gblk, 0, stream>>>(sim, dis_W1, dis_b1, e1, HID, VOCAB);
    gemm(e1, dis_W2, dis_b2, e2, MROWS, HID, HID, 1, 0, stream);
    gemm(e2, dis_Wih, dis_bih, xw3, MROWS, 3 * HID, HID, 0, 0, stream);
    hipMemsetAsync(hA, 0, BATCH * HID * sizeof(float), stream);
    {
        float* hc = hA; float* hn = hB;
        for (int s = 0; s < S_LEN; ++s) {
            gemm(hc, dis_Whh, dis_bhh, hg, BATCH, 3 * HID, HID, 0, 0, stream);
            gru_gate<<<gate_blocks, 256, 0, stream>>>(xw3 + (size_t)s * BATCH * 3 * HID, hg, hc, hn, HID);
            float* t = hc; hc = hn; hn = t;
        }
        gemm(hc, dis_lin_W, dis_lin_b, enc_sim_dst, BATCH, HID, HID, 0, 0, stream);
    }

    // ============ discriminator on softmax(out) (exp fused into A path) ============
    gemm(out, dis_W1, dis_b1, e1, MROWS, HID, VOCAB, 1, 1, stream);  // g1 = thresh(exp(out)@W1^T+b)
    gemm(e1, dis_W2, dis_b2, e2, MROWS, HID, HID, 1, 0, stream);
    gemm(e2, dis_Wih, dis_bih, xw3, MROWS, 3 * HID, HID, 0, 0, stream);
    hipMemsetAsync(hA, 0, BATCH * HID * sizeof(float), stream);
    {
        float* hc = hA; float* hn = hB;
        for (int s = 0; s < S_LEN; ++s) {
            gemm(hc, dis_Whh, dis_bhh, hg, BATCH, 3 * HID, HID, 0, 0, stream);
            gru_gate<<<gate_blocks, 256, 0, stream>>>(xw3 + (size_t)s * BATCH * 3 * HID, hg, hc, hn, HID);
            float* t = hc; hc = hn; hn = t;
        }
        gemm(hc, dis_lin_W, dis_lin_b, enc_out_dst, BATCH, HID, HID, 0, 0, stream);
    }
}